// GCNAggregator_67757404061981
// MI455X (gfx1250) — compile-verified
//
#include <hip/hip_runtime.h>
#include <hip/hip_bf16.h>
#include <stdint.h>

// ---------------------------------------------------------------------------
// GCN mean-aggregator for MI455X (gfx1250).
// Pure L2-gather-bound problem (0.25 FLOP/byte; 51 MB table fits in 192 MB L2)
// -> CDNA5 async-copy path: GLOBAL_LOAD_ASYNC_TO_LDS_B128 (ASYNCcnt) keeps 32
//    random 512B row-gathers in flight per wave with zero VGPR landing zone;
//    staged s_wait_asynccnt overlaps the LDS read-back + VALU reduction with
//    the gather tail. Indices are broadcast via v_readlane (uniform/SGPR) so
//    the LDS pipe is reserved for the staging reads. WMMA deliberately NOT
//    used: the gather structure is node-private (block-diagonal mask), so
//    matrix ops would be 16x wasted work while the bottleneck stays in the
//    gather path.
// (Round-4 note: carrying the per-row LDS address in IOFFSET was tried and
//  reverted — IOFFSET adds to both LDS and global addresses, forcing 32
//  s_mov_b64 compensation constants; per-row LDS address in a VGPR is cheaper.)
// ---------------------------------------------------------------------------

#define FEAT_DIM 128          // floats per row (512 bytes)
#define KNEIGH   32           // sampled neighbors per node (== wave32 lanes)
#define WAVES_PER_BLOCK 2
#define BLOCK_THREADS (WAVES_PER_BLOCK * 32)

typedef int v4i __attribute__((ext_vector_type(4)));
typedef __attribute__((address_space(1))) v4i* g_v4i_ptr;   // global int4*
typedef __attribute__((address_space(3))) v4i* l_v4i_ptr;   // LDS int4*

#if defined(__HIP_DEVICE_COMPILE__)
#if __has_builtin(__builtin_amdgcn_global_load_async_to_lds_b128)
#define HAVE_ASYNC_BUILTIN 1
#else
#define HAVE_ASYNC_BUILTIN 0
#endif
#if __has_builtin(__builtin_amdgcn_s_wait_asynccnt)
#define HAVE_WAIT_ASYNC_BUILTIN 1
#else
#define HAVE_WAIT_ASYNC_BUILTIN 0
#endif
#endif

// Wave-wide async copy: each lane moves 16 bytes global -> LDS.
// One instruction == one 512-byte feature row for the whole wave.
__device__ __forceinline__ void async_row16(const float* gsrc, float* lds_dst) {
#if defined(__HIP_DEVICE_COMPILE__)
#if HAVE_ASYNC_BUILTIN
    __builtin_amdgcn_global_load_async_to_lds_b128(
        (g_v4i_ptr)gsrc, (l_v4i_ptr)lds_dst, /*offset=*/0, /*cpol=*/0);
#else
    unsigned lds_off = (unsigned)(unsigned long long)(l_v4i_ptr)lds_dst;
    unsigned long long gaddr = (unsigned long long)gsrc;
    asm volatile("global_load_async_to_lds_b128 %0, %1, off"
                 :: "v"(lds_off), "v"(gaddr)
                 : "memory");
#endif
#endif
}

// Wait until at most N async ops remain outstanding (ICE via template).
template <int N>
__device__ __forceinline__ void wait_async_le() {
#if defined(__HIP_DEVICE_COMPILE__)
#if HAVE_WAIT_ASYNC_BUILTIN
    __builtin_amdgcn_s_wait_asynccnt(N);
#else
    asm volatile("s_wait_asynccnt %0" :: "n"(N) : "memory");
#endif
#endif
}

// Accumulate 8 staged rows into the 4 per-lane accumulators.
__device__ __forceinline__ void acc8(const float* stage, int c0, int j0,
                                     float& a0, float& a1, float& a2, float& a3) {
    #pragma unroll
    for (int jj = 0; jj < 8; ++jj) {
        const float4 v = *(const float4*)(stage + (j0 + jj) * FEAT_DIM + c0);
        a0 += v.x; a1 += v.y; a2 += v.z; a3 += v.w;
    }
}

__global__ __launch_bounds__(BLOCK_THREADS)
void GCNAggregator_kernel(const float* __restrict__ features,
                          const int*   __restrict__ nodes,
                          const int*   __restrict__ neigh,
                          float*       __restrict__ out,
                          int B, int K, float inv)
{
    // 2 waves * 32 rows * 512B = 32 KB static LDS -> ~10 blocks per WGP.
    __shared__ float smem[WAVES_PER_BLOCK * KNEIGH * FEAT_DIM];

    const int lane      = threadIdx.x & 31;
    const int waveInBlk = threadIdx.x >> 5;
    const int node      = blockIdx.x * WAVES_PER_BLOCK + waveInBlk;
    if (node >= B) return;

    float* stage = smem + waveInBlk * (KNEIGH * FEAT_DIM);
    const int c0 = lane * 4;                 // this lane's 4 feature columns

    // Lane l holds neighbor index l (K == 32 == wave width).
    const int nb = neigh[(size_t)node * K + lane];

    // Issue all 32 row gathers asynchronously (ASYNCcnt <= 32 < 63 cap).
    // readlane -> index in SGPR: scalar address math, LDS pipe left free.
    #pragma unroll
    for (int j = 0; j < KNEIGH; ++j) {
        const int idx = __builtin_amdgcn_readlane(nb, j);
        const float* src = features + (size_t)(unsigned)idx * FEAT_DIM + c0;
        async_row16(src, stage + j * FEAT_DIM + c0);
    }

    // Self row via a normal b128 load; overlaps the in-flight async stream.
    const int root = nodes[node];
    const float4 r = *(const float4*)(features + (size_t)(unsigned)root * FEAT_DIM + c0);
    float a0 = r.x, a1 = r.y, a2 = r.z, a3 = r.w;

    // Async loads complete in order: consume in groups of 8 while the
    // remaining gathers are still in flight.
    wait_async_le<24>(); acc8(stage, c0,  0, a0, a1, a2, a3);
    wait_async_le<16>(); acc8(stage, c0,  8, a0, a1, a2, a3);
    wait_async_le<8>();  acc8(stage, c0, 16, a0, a1, a2, a3);
    wait_async_le<0>();  acc8(stage, c0, 24, a0, a1, a2, a3);

    float4 o;
    o.x = a0 * inv; o.y = a1 * inv; o.z = a2 * inv; o.w = a3 * inv;
    *(float4*)(out + (size_t)node * FEAT_DIM + c0) = o;
}

extern "C" void kernel_launch(void* const* d_in, const int* in_sizes, int n_in,
                              void* d_out, int out_size, void* d_ws, size_t ws_size,
                              hipStream_t stream) {
    (void)n_in; (void)out_size; (void)d_ws; (void)ws_size;

    const float* features = (const float*)d_in[0];   // [V, 128] fp32
    const int*   nodes    = (const int*)d_in[1];     // [B] int
    const int*   neigh    = (const int*)d_in[2];     // [B, K] int
    float*       out      = (float*)d_out;           // [B, 128] fp32

    const int B = in_sizes[1];
    const int K = (B > 0) ? (in_sizes[2] / B) : KNEIGH;   // == 32
    const float inv = 1.0f / (float)(K + 1);

    const int blocks = (B + WAVES_PER_BLOCK - 1) / WAVES_PER_BLOCK;
    GCNAggregator_kernel<<<dim3(blocks), dim3(BLOCK_THREADS), 0, stream>>>(
        features, nodes, neigh, out, B, K, inv);
}